// CustomLoss_87952340287807
// MI455X (gfx1250) — compile-verified
//
#include <hip/hip_runtime.h>
#include <hip/hip_bf16.h>

typedef __attribute__((ext_vector_type(2))) float v2f;
typedef __attribute__((ext_vector_type(4))) float v4f;
typedef __attribute__((ext_vector_type(8))) float v8f;

// Hardware v_sin_f32/v_cos_f32 compute sin/cos(2*pi * x). The reference's
// angle is exactly 2*pi*(v - 0.5), so we pass (v - 0.5) directly: one trans
// instruction per value, argument already in the valid [-0.5, 0.5) domain.
__device__ __forceinline__ void euler_vec(float ry, float rp, float rr,
                                          float& v0, float& v1, float& v2) {
    float cy = __builtin_amdgcn_cosf(ry), sy = __builtin_amdgcn_sinf(ry);
    float cp = __builtin_amdgcn_cosf(rp), sp = __builtin_amdgcn_sinf(rp);
    float cr = __builtin_amdgcn_cosf(rr), sr = __builtin_amdgcn_sinf(rr);
    float spcr = sp * cr;
    v0 = cy * spcr + sy * sr;
    v1 = sy * spcr - cy * sr;
    v2 = cp * cr;
}

__device__ __forceinline__ float per_sample(float oy, float op, float orr,
                                            float ty, float tp, float tr) {
    float a0, a1, a2, b0, b1, b2;
    euler_vec(oy - 0.5f, op - 0.5f, orr - 0.5f, a0, a1, a2);
    euler_vec(ty - 0.5f, tp - 0.5f, tr - 0.5f, b0, b1, b2);
    float c0 = a1 * b2 - a2 * b1;
    float c1 = a2 * b0 - a0 * b2;
    float c2 = a0 * b1 - a1 * b0;
    float cn = __builtin_sqrtf(c0 * c0 + c1 * c1 + c2 * c2);
    float d  = a0 * b0 + a1 * b1 + a2 * b2;
    return 1.0f - cn + d;
}

// Sum of one 4-sample group starting at sample s (s+3 < n guaranteed).
// 3x b128 non-temporal loads per tensor: streamed once, no reuse -> TH=NT.
__device__ __forceinline__ float group4(const float* __restrict__ out,
                                        const float* __restrict__ tgt, long s) {
    const v4f* o4 = (const v4f*)(out + s * 3);
    const v4f* t4 = (const v4f*)(tgt + s * 3);
    v4f oa = __builtin_nontemporal_load(o4);
    v4f ob = __builtin_nontemporal_load(o4 + 1);
    v4f oc = __builtin_nontemporal_load(o4 + 2);
    v4f ta = __builtin_nontemporal_load(t4);
    v4f tb = __builtin_nontemporal_load(t4 + 1);
    v4f tc = __builtin_nontemporal_load(t4 + 2);
    float s0 = per_sample(oa.x, oa.y, oa.z, ta.x, ta.y, ta.z);
    float s1 = per_sample(oa.w, ob.x, ob.y, ta.w, tb.x, tb.y);
    float s2 = per_sample(ob.z, ob.w, oc.x, tb.z, tb.w, tc.x);
    float s3 = per_sample(oc.y, oc.z, oc.w, tc.y, tc.z, tc.w);
    return (s0 + s1) + (s2 + s3);
}

// Main streaming kernel: 16 samples per thread (4 groups of 4), per-wave shfl
// reduction, per-block LDS reduction -> one partial per block (2048 blocks).
__global__ __launch_bounds__(256) void loss_main(const float* __restrict__ out,
                                                 const float* __restrict__ tgt,
                                                 float* __restrict__ block_sums,
                                                 int n) {
    const int gid = blockIdx.x * blockDim.x + threadIdx.x;
    const long s0 = (long)gid * 16;         // first sample of this thread
    float sum = 0.0f;

    if (s0 + 15 < (long)n) {                // fast path: 192B per tensor
        #pragma unroll
        for (int g = 0; g < 4; ++g)
            sum += group4(out, tgt, s0 + 4 * g);
    } else {                                // tail (unused for B = 8M, kept safe)
        for (long s = s0; s < (long)n && s < s0 + 16; ++s) {
            const float* o = out + s * 3;
            const float* t = tgt + s * 3;
            sum += per_sample(o[0], o[1], o[2], t[0], t[1], t[2]);
        }
    }

    // wave32 tree reduction
    #pragma unroll
    for (int off = 16; off > 0; off >>= 1)
        sum += __shfl_down(sum, off, 32);

    __shared__ float lds[8];                // 256 threads = 8 wave32s
    if ((threadIdx.x & 31) == 0) lds[threadIdx.x >> 5] = sum;
    __syncthreads();
    if (threadIdx.x == 0) {
        float t = 0.0f;
        #pragma unroll
        for (int w = 0; w < 8; ++w) t += lds[w];
        block_sums[blockIdx.x] = t;
    }
}

// Zero-fill padding region of the partials array (only launched when the
// block count is not a multiple of 512; no-op for B = 8388608).
__global__ void pad_zero(float* __restrict__ p, int from, int to) {
    int i = from + blockIdx.x * blockDim.x + threadIdx.x;
    if (i < to) p[i] = 0.0f;
}

// Final deterministic reduction on one wave32 using v_wmma_f32_16x16x4_f32
// with B = ones. count is a multiple of 512 (padded with zeros), so all loads
// are unconditional aligned b64: 8 loads in flight per outer iteration, then
// 8 accumulating WMMAs -> only count/512 memory round-trips on this wave.
__global__ __launch_bounds__(32) void final_reduce(const float* __restrict__ partials,
                                                   int count,
                                                   float* __restrict__ result) {
    const int lane  = threadIdx.x;          // 0..31, EXEC all ones for WMMA
    const int m     = lane & 15;
    const int kbase = (lane >> 4) * 2;      // lanes 0-15: K=0,1 ; lanes 16-31: K=2,3
    const int loff  = 4 * m + kbase;        // this lane's offset within a 64-chunk

    v2f ones; ones[0] = 1.0f; ones[1] = 1.0f;
    v8f acc = {};

    for (int base = 0; base < count; base += 512) {
        v2f a[8];
        #pragma unroll
        for (int u = 0; u < 8; ++u)         // issue all 8 b64 loads back-to-back
            a[u] = *(const v2f*)(partials + base + 64 * u + loff);
        #pragma unroll
        for (int u = 0; u < 8; ++u)
            // 8 args: (neg_a, A, neg_b, B, c_mod, C, reuse_a, reuse_b)
            acc = __builtin_amdgcn_wmma_f32_16x16x4_f32(
                false, a[u], false, ones, (short)0, acc, false, false);
    }

    // D layout: VGPR j -> M=j (lanes 0-15) / M=j+8 (lanes 16-31), N = lane%16.
    // Every column N is identical, so lane0 (rows 0-7) + lane16 (rows 8-15)
    // at column 0 gives the full sum.
    float s = acc[0] + acc[1] + acc[2] + acc[3] + acc[4] + acc[5] + acc[6] + acc[7];
    float other = __shfl(s, lane ^ 16, 32);
    if (lane == 0) result[0] = s + other;
}

extern "C" void kernel_launch(void* const* d_in, const int* in_sizes, int n_in,
                              void* d_out, int out_size, void* d_ws, size_t ws_size,
                              hipStream_t stream) {
    const float* out_p = (const float*)d_in[0];
    const float* tgt_p = (const float*)d_in[1];
    const int n = in_sizes[0] / 3;          // samples

    const int BS = 256, SPT = 16;
    const int threads = (n + SPT - 1) / SPT;
    const int blocks  = (threads + BS - 1) / BS;       // 2048 for B = 8388608
    const int padded  = ((blocks + 511) / 512) * 512;  // multiple of 512

    float* partials = (float*)d_ws;                    // padded * 4 bytes

    if (padded > blocks) {
        const int extra = padded - blocks;
        pad_zero<<<(extra + 255) / 256, 256, 0, stream>>>(partials, blocks, padded);
    }
    loss_main<<<blocks, BS, 0, stream>>>(out_p, tgt_p, partials, n);
    final_reduce<<<1, 32, 0, stream>>>(partials, padded, (float*)d_out);
}